// ECE_36481452212818
// MI455X (gfx1250) — compile-verified
//
#include <hip/hip_runtime.h>

typedef float v4f __attribute__((ext_vector_type(4)));
typedef int   v4i __attribute__((ext_vector_type(4)));
typedef float v2f __attribute__((ext_vector_type(2)));
typedef float v8f __attribute__((ext_vector_type(8)));

#define NBINS 15
#define TPB   256
#define WAVES (TPB / 32)           // 8 waves per block (wave32)
#define REPL  4                    // histogram replicas per wave (conflict mitigation)
#define HROWS (WAVES * REPL)       // 32 rows of 16 bins in LDS
#define HSIZE (HROWS * 16)         // 512 floats

// ---------------------------------------------------------------------------
// Main streaming kernel: grid-stride over float4/int4, per-wave replicated
// LDS histograms via ds_add_f32, block reduction of the 32x16 histogram with
// 8 chained V_WMMA_F32_16X16X4_F32 (A = ones => D[m][n] = sum_k B[k][n]).
// ---------------------------------------------------------------------------
__global__ void __launch_bounds__(TPB) ece_main_kernel(
    const float* __restrict__ logits,
    const int* __restrict__ labels,
    float* __restrict__ part,          // [gridDim.x][16] partial bin sums
    unsigned n)
{
    __shared__ float hist[HSIZE];
    for (int i = threadIdx.x; i < HSIZE; i += TPB) hist[i] = 0.0f;
    __syncthreads();

    const int wave  = threadIdx.x >> 5;
    const int lane  = threadIdx.x & 31;
    const int hbase = wave * (REPL * 16) + (lane & (REPL - 1)) * 16;

    // Per-element accumulate: sigmoid, accuracy, bin, LDS float atomic.
    auto accum = [&](float x, int lab) {
        float p = __builtin_amdgcn_rcpf(1.0f + __expf(-x));   // sigmoid
        int   predi = (p > 0.5f) ? 1 : 0;
        float accf  = (predi == lab) ? 1.0f : 0.0f;
        float contrib = (p > 0.0f) ? (p - accf) : 0.0f;       // valid mask
        int bin = (int)ceilf(p * 15.0f) - 1;                  // digitize right=True
        bin = bin < 0 ? 0 : (bin > (NBINS - 1) ? (NBINS - 1) : bin);
        atomicAdd(&hist[hbase + bin], contrib);               // ds_add_f32
    };

    const unsigned n4     = n >> 2;
    const unsigned stride = gridDim.x * TPB;
    unsigned       idx    = blockIdx.x * TPB + threadIdx.x;
    const v4f* lg4 = (const v4f*)logits;
    const v4i* lb4 = (const v4i*)labels;

    for (; idx < n4; idx += stride) {
        if (idx + stride < n4) {                      // speculative prefetch next tile
            __builtin_prefetch(&lg4[idx + stride], 0, 0);   // global_prefetch_b8
            __builtin_prefetch(&lb4[idx + stride], 0, 0);
        }
        v4f x = __builtin_nontemporal_load(&lg4[idx]);      // B128 load, TH=NT
        v4i l = __builtin_nontemporal_load(&lb4[idx]);      // B128 load, TH=NT
        accum(x.x, l.x);
        accum(x.y, l.y);
        accum(x.z, l.z);
        accum(x.w, l.w);
    }

    // Scalar tail (n not multiple of 4) handled by block 0.
    if (blockIdx.x == 0) {
        for (unsigned i = (n4 << 2) + threadIdx.x; i < n; i += TPB)
            accum(logits[i], labels[i]);
    }
    __syncthreads();

    // ---- Block reduction of 32x16 histogram via WMMA -----------------------
    // All 8 waves execute with EXEC all-ones (WMMA requirement); each wave
    // redundantly reduces the full histogram; wave 0 writes the result.
    // A = ones(16x4): D[m][n] = sum_k B[k][n] + C[m][n], independent of the
    // K-slot ordering inside B. B slots: half = lane>>4, col = lane&15; the
    // two B VGPRs of WMMA i cover rows {4i+half, 4i+2+half} for half in {0,1},
    // i.e. all 4 rows of group i exactly once.
    {
        const int half = lane >> 4;
        const int col  = lane & 15;
        v2f a; a.x = 1.0f; a.y = 1.0f;
        v8f acc = {};
#pragma unroll
        for (int i = 0; i < 8; ++i) {
            v2f b;
            b.x = hist[(4 * i + half) * 16 + col];
            b.y = hist[(4 * i + 2 + half) * 16 + col];
            acc = __builtin_amdgcn_wmma_f32_16x16x4_f32(
                false, a, false, b, (short)0, acc, false, false);
        }
        // D layout: VGPR0, lanes 0..15 hold row M=0, N=lane -> bin sums.
        if (threadIdx.x < NBINS)
            part[(unsigned)blockIdx.x * 16u + threadIdx.x] = acc[0];
    }
}

// ---------------------------------------------------------------------------
// Final reduction: sum 15 bins over all block partials, ece = sum|S_b| / N.
// Deterministic (fixed summation order, no atomics).
// ---------------------------------------------------------------------------
__global__ void __launch_bounds__(256) ece_final_kernel(
    const float* __restrict__ part, float* __restrict__ out,
    int nPart, float invN)
{
    __shared__ float red[256];
    float ece = 0.0f;
    for (int b = 0; b < NBINS; ++b) {
        float s = 0.0f;
        for (int i = threadIdx.x; i < nPart; i += 256)
            s += part[i * 16 + b];
        red[threadIdx.x] = s;
        __syncthreads();
        for (int off = 128; off > 0; off >>= 1) {
            if ((int)threadIdx.x < off) red[threadIdx.x] += red[threadIdx.x + off];
            __syncthreads();
        }
        if (threadIdx.x == 0) ece += fabsf(red[0]);
        __syncthreads();
    }
    if (threadIdx.x == 0) out[0] = ece * invN;
}

extern "C" void kernel_launch(void* const* d_in, const int* in_sizes, int n_in,
                              void* d_out, int out_size, void* d_ws, size_t ws_size,
                              hipStream_t stream)
{
    const float* logits = (const float*)d_in[0];
    const int*   labels = (const int*)d_in[1];
    unsigned n = (unsigned)in_sizes[0];

    float* part = (float*)d_ws;
    // 2048 persistent blocks: 16 blocks/WGP-class occupancy, keeps the
    // per-bin partial count small; clamp to available workspace.
    int blocks = 2048;
    int maxBlocks = (int)(ws_size / (16 * sizeof(float)));
    if (blocks > maxBlocks) blocks = maxBlocks;
    if (blocks < 1) blocks = 1;

    ece_main_kernel<<<blocks, TPB, 0, stream>>>(logits, labels, part, n);
    ece_final_kernel<<<1, 256, 0, stream>>>(part, (float*)d_out, blocks,
                                            1.0f / (float)n);
}